// RealSphericalHarmonicTransform_53523882443328
// MI455X (gfx1250) — compile-verified
//
#include <hip/hip_runtime.h>
#include <hip/hip_bf16.h>

typedef __attribute__((ext_vector_type(2))) float v2f;
typedef __attribute__((ext_vector_type(8))) float v8f;

#define N_THETA   360
#define N_LAMBDA  720
#define M_MAX     361      // rfft modes
#define NCOLS     722      // 2*M_MAX columns (re, im interleaved)
#define NCOLS_PAD 736      // 46 tiles of 16
#define NH        8        // batch*heads rows per (theta)
#define NROWS     2880     // NH * N_THETA
#define MTILES1   180      // NROWS/16
#define NTILES1   46       // NCOLS_PAD/16
#define NTILES2   23       // ceil(360/16)

// ---------------------------------------------------------------------------
// Kernel 0: build the scaled Fourier matrix F[l][c], c = 2m (+0: cos, +1: -sin),
// scale = 2*pi/720 (forward-normalized rfft times 2*pi). Padded cols are zero.
// ---------------------------------------------------------------------------
__global__ void build_F(float* __restrict__ F) {
    int idx = blockIdx.x * blockDim.x + threadIdx.x;
    if (idx >= N_LAMBDA * NCOLS_PAD) return;
    int l = idx / NCOLS_PAD;
    int c = idx - l * NCOLS_PAD;
    float val = 0.0f;
    if (c < NCOLS) {
        int m = c >> 1;
        int phase = (m * l) % N_LAMBDA;            // exact integer reduction
        const float w0 = 6.2831853071795864769f / (float)N_LAMBDA; // 2*pi/720
        float ang = w0 * (float)phase;
        float s, co;
        __sincosf(ang, &s, &co);
        val = (c & 1) ? (-w0 * s) : (w0 * co);
    }
    F[idx] = val;
}

// ---------------------------------------------------------------------------
// Kernel 1: DFT as GEMM with V_WMMA_F32_16X16X4_F32.
//   A = x reshaped [2880, 720], B = F [720, 736], one wave per 16x16 D tile.
// D is scatter-stored into Xws[m][k][16] with fast index = {re h0..7, im h0..7}.
// ---------------------------------------------------------------------------
__global__ void dft_gemm(const float* __restrict__ x, const float* __restrict__ F,
                         float* __restrict__ Xws) {
    int wave = blockIdx.x * 8 + (threadIdx.x >> 5);
    int lane = threadIdx.x & 31;
    if (wave >= MTILES1 * NTILES1) return;        // wave-uniform: EXEC stays full
    int mt = wave / NTILES1;
    int nt = wave - mt * NTILES1;
    int r0 = mt * 16;
    int c0 = nt * 16;
    int lm = lane & 15;                            // row (A) / col (B,D) in tile
    int lk = (lane >> 4) << 1;                     // K sub-offset: 0 or 2

    const float* Arow = x + (size_t)(r0 + lm) * N_LAMBDA;  // A: row r0+lm
    const float* Bcol = F + c0 + lm;                       // B: col c0+lm

    v8f acc = {};
    for (int k0 = 0; k0 < N_LAMBDA; k0 += 4) {
        v2f a = *(const v2f*)(Arow + k0 + lk);             // K = k0+lk, k0+lk+1
        v2f b;
        b.x = Bcol[(size_t)(k0 + lk) * NCOLS_PAD];
        b.y = Bcol[(size_t)(k0 + lk + 1) * NCOLS_PAD];
        acc = __builtin_amdgcn_wmma_f32_16x16x4_f32(
            false, a, false, b, (short)0, acc, false, false);
    }

    int c = c0 + lm;                               // D col held by this lane
    if (c < NCOLS) {
        int m  = c >> 1;
        int im = c & 1;
        int rbase = r0 + ((lane >> 4) << 3);       // D rows: v + 8*(lane/16)
        float vals[8];
        *(v8f*)vals = acc;
        #pragma unroll
        for (int v = 0; v < 8; ++v) {
            int r = rbase + v;                     // global row = h*360 + k
            int h = r / N_THETA;
            int k = r - h * N_THETA;
            Xws[((size_t)m * N_THETA + k) * 16 + im * 8 + h] = vals[v];
        }
    }
}

// ---------------------------------------------------------------------------
// Kernel 2: per-mode Legendre contraction. For each (m, n-tile):
//   A[16 x 360] = Xws[m][k][0..15]  (rows 0-7 = re(h), rows 8-15 = im(h))
//   B[360 x 16] = w[m][n][k] transposed on the fly
//   D[16 x 16]  -> out[h][n][m] complex64 interleaved
// ---------------------------------------------------------------------------
__global__ void legendre_gemm(const float* __restrict__ Xws, const float* __restrict__ w,
                              float* __restrict__ out) {
    int wave = blockIdx.x * 8 + (threadIdx.x >> 5);
    int lane = threadIdx.x & 31;
    if (wave >= M_MAX * NTILES2) return;           // wave-uniform
    int m  = wave / NTILES2;
    int nt = wave - m * NTILES2;
    int n0 = nt * 16;
    int lm = lane & 15;
    int lk = (lane >> 4) << 1;

    int n  = n0 + lm;
    int nc = (n < N_THETA) ? n : (N_THETA - 1);    // clamp: junk cols never stored

    const float* Xm   = Xws + (size_t)m * (N_THETA * 16);
    const float* Wrow = w + ((size_t)m * N_THETA + nc) * N_THETA;  // w[m][nc][*]

    v8f acc = {};
    for (int k0 = 0; k0 < N_THETA; k0 += 4) {
        __builtin_prefetch(Wrow + k0 + 64, 0, 1);  // global_prefetch_b8
        v2f a;
        a.x = Xm[(size_t)(k0 + lk) * 16 + lm];
        a.y = Xm[(size_t)(k0 + lk + 1) * 16 + lm];
        v2f b = *(const v2f*)(Wrow + k0 + lk);
        acc = __builtin_amdgcn_wmma_f32_16x16x4_f32(
            false, a, false, b, (short)0, acc, false, false);
    }

    if (n < N_THETA) {
        float vals[8];
        *(v8f*)vals = acc;
        int g = (lane >> 4) << 3;                  // 0 or 8
        #pragma unroll
        for (int v = 0; v < 8; ++v) {
            int M   = g + v;                       // 0-7 -> re, 8-15 -> im
            int h   = M & 7;
            int imf = M >> 3;
            out[2 * (((size_t)h * N_THETA + n) * M_MAX + m) + imf] = vals[v];
        }
    }
}

// ---------------------------------------------------------------------------
extern "C" void kernel_launch(void* const* d_in, const int* in_sizes, int n_in,
                              void* d_out, int out_size, void* d_ws, size_t ws_size,
                              hipStream_t stream) {
    const float* x = (const float*)d_in[0];   // [1,8,360,720]
    const float* w = (const float*)d_in[1];   // [361,360,360]
    float* out = (float*)d_out;               // [1,8,360,361] complex64 interleaved

    float* F   = (float*)d_ws;                // 720*736 floats
    float* Xws = F + N_LAMBDA * NCOLS_PAD;    // 361*360*16 floats

    build_F<<<(N_LAMBDA * NCOLS_PAD + 255) / 256, 256, 0, stream>>>(F);

    int waves1 = MTILES1 * NTILES1;           // 8280
    dft_gemm<<<(waves1 + 7) / 8, 256, 0, stream>>>(x, F, Xws);

    int waves2 = M_MAX * NTILES2;             // 8303
    legendre_gemm<<<(waves2 + 7) / 8, 256, 0, stream>>>(Xws, w, out);
}